// PSRoIPooling_17188459118946
// MI455X (gfx1250) — compile-verified
//
#include <hip/hip_runtime.h>
#include <cstdint>

// PS-ROI pooling (crop_and_resize style) for MI455X / gfx1250.
// inputs: [8,64,64,2304] f32, boxes: [8,128,4] f32 -> out [8,128,256] f32.
// Pipelined GLOBAL_LOAD_ASYNC_TO_LDS_B128 gather, depth-4 double buffering.

#define BATCH 8
#define NBOX  128
#define HH    64
#define WW    64
#define CH    2304
#define SS    256      // per-bin channels
#define NSAMP 81       // 3x3 bins * 3x3 crop points
#define GPB   4        // boxes (groups) per block
#define LPG   64       // lanes per box group (each lane = 4 channels)
#define DEPTH 4        // samples in flight (parity = k & 3)

__global__ __launch_bounds__(256)
void psroi_kernel(const float* __restrict__ inp,
                  const float* __restrict__ boxes,
                  float* __restrict__ out)
{
    __shared__ float sw0[GPB][NSAMP], sw1[GPB][NSAMP], sw2[GPB][NSAMP], sw3[GPB][NSAMP];
    __shared__ int   so0[GPB][NSAMP], so1[GPB][NSAMP], so2[GPB][NSAMP], so3[GPB][NSAMP];
    __shared__ float4 dbuf[DEPTH][GPB][4][LPG];   // [parity][group][tap][lane]

    const int tid = threadIdx.x;          // 0..255
    const int blk = blockIdx.x;           // 0..255
    const int b   = blk >> 5;             // batch
    const int nb  = (blk & 31) << 2;      // first of 4 boxes handled here

    // ---- per-(box,sample) metadata: 4 boxes x 81 samples ----
    for (int idx = tid; idx < GPB * NSAMP; idx += 256) {
        const int g = idx / NSAMP;
        const int k = idx - g * NSAMP;
        const int n = nb + g;

        const int bidx = (b * NBOX + n) * 4;
        const float y1 = boxes[bidx + 0];
        const float x1 = boxes[bidx + 1];
        const float y2 = boxes[bidx + 2];
        const float x2 = boxes[bidx + 3];
        const float sy = (y2 - y1) * (1.0f / 3.0f);
        const float sx = (x2 - x1) * (1.0f / 3.0f);

        const int j  = k % 3;             // x crop index
        const int i  = (k / 3) % 3;       // y crop index
        const int bx = (k / 9) % 3;
        const int by = k / 27;
        const int bin = by * 3 + bx;

        // ys = (H-1)*(y1 + by*sy + i*sy/2), H-1 = 63
        const float ys = 63.0f * (y1 + (float)by * sy + (float)i * sy * 0.5f);
        const float xs = 63.0f * (x1 + (float)bx * sx + (float)j * sx * 0.5f);
        const bool vy = (ys >= 0.0f) && (ys <= 63.0f);
        const bool vx = (xs >= 0.0f) && (xs <= 63.0f);

        const float ysc = fminf(fmaxf(ys, 0.0f), 63.0f);
        const float xsc = fminf(fmaxf(xs, 0.0f), 63.0f);
        const float y0f = floorf(ysc);
        const float x0f = floorf(xsc);
        const float ly  = ysc - y0f;
        const float lx  = xsc - x0f;
        const int y0i = (int)y0f;
        const int x0i = (int)x0f;
        const int y1i = min(y0i + 1, 63);
        const int x1i = min(x0i + 1, 63);

        const float wv = (vy && vx) ? (1.0f / (float)NSAMP) : 0.0f;
        sw0[g][k] = wv * (1.0f - ly) * (1.0f - lx);
        sw1[g][k] = wv * (1.0f - ly) * lx;
        sw2[g][k] = wv * ly * (1.0f - lx);
        sw3[g][k] = wv * ly * lx;

        const int base = b * (HH * WW * CH);
        so0[g][k] = base + ((y0i * WW + x0i) * 9 + bin) * SS;
        so1[g][k] = base + ((y0i * WW + x1i) * 9 + bin) * SS;
        so2[g][k] = base + ((y1i * WW + x0i) * 9 + bin) * SS;
        so3[g][k] = base + ((y1i * WW + x1i) * 9 + bin) * SS;
    }
    __syncthreads();

    const int g = tid >> 6;               // box group 0..3
    const int l = tid & 63;               // lane in group
    const int c = l << 2;                 // first of 4 channels

    // LDS byte address: low 32 bits of flat shared pointer == LDS offset.
    const uint32_t ldsb = (uint32_t)(uintptr_t)(&dbuf[0][g][0][l]);
    const uint32_t PAR_STRIDE = (uint32_t)(sizeof(float4) * GPB * 4 * LPG);  // 16 KB
    const uint32_t TAP_STRIDE = (uint32_t)(sizeof(float4) * LPG);           // 1 KB

    auto issue = [&](int k) {
        const uint32_t lb = ldsb + (uint32_t)(k & 3) * PAR_STRIDE;
        const float* p0 = inp + so0[g][k] + c;
        const float* p1 = inp + so1[g][k] + c;
        const float* p2 = inp + so2[g][k] + c;
        const float* p3 = inp + so3[g][k] + c;
        asm volatile("global_load_async_to_lds_b128 %0, %1, off"
                     :: "v"(lb + 0u * TAP_STRIDE), "v"(p0) : "memory");
        asm volatile("global_load_async_to_lds_b128 %0, %1, off"
                     :: "v"(lb + 1u * TAP_STRIDE), "v"(p1) : "memory");
        asm volatile("global_load_async_to_lds_b128 %0, %1, off"
                     :: "v"(lb + 2u * TAP_STRIDE), "v"(p2) : "memory");
        asm volatile("global_load_async_to_lds_b128 %0, %1, off"
                     :: "v"(lb + 3u * TAP_STRIDE), "v"(p3) : "memory");
    };

    float4 acc = make_float4(0.0f, 0.0f, 0.0f, 0.0f);

    auto consume = [&](int k) {
        const int par = k & 3;
        const float w0 = sw0[g][k];
        const float w1 = sw1[g][k];
        const float w2 = sw2[g][k];
        const float w3 = sw3[g][k];
        const float4 v0 = dbuf[par][g][0][l];
        const float4 v1 = dbuf[par][g][1][l];
        const float4 v2 = dbuf[par][g][2][l];
        const float4 v3 = dbuf[par][g][3][l];
        acc.x = fmaf(w0, v0.x, acc.x); acc.y = fmaf(w0, v0.y, acc.y);
        acc.z = fmaf(w0, v0.z, acc.z); acc.w = fmaf(w0, v0.w, acc.w);
        acc.x = fmaf(w1, v1.x, acc.x); acc.y = fmaf(w1, v1.y, acc.y);
        acc.z = fmaf(w1, v1.z, acc.z); acc.w = fmaf(w1, v1.w, acc.w);
        acc.x = fmaf(w2, v2.x, acc.x); acc.y = fmaf(w2, v2.y, acc.y);
        acc.z = fmaf(w2, v2.z, acc.z); acc.w = fmaf(w2, v2.w, acc.w);
        acc.x = fmaf(w3, v3.x, acc.x); acc.y = fmaf(w3, v3.y, acc.y);
        acc.z = fmaf(w3, v3.z, acc.z); acc.w = fmaf(w3, v3.w, acc.w);
    };

    // ---- depth-4 software pipeline: 3 samples in flight ahead of consume ----
    issue(0); issue(1); issue(2);

    #pragma unroll 1
    for (int k = 0; k < NSAMP - 3; ++k) {
        issue(k + 3);
        // 16 outstanding; retire the oldest sample's 4 loads
        asm volatile("s_wait_asynccnt 0xc" ::: "memory");
        consume(k);
    }
    asm volatile("s_wait_asynccnt 0x8" ::: "memory");
    consume(NSAMP - 3);
    asm volatile("s_wait_asynccnt 0x4" ::: "memory");
    consume(NSAMP - 2);
    asm volatile("s_wait_asynccnt 0x0" ::: "memory");
    consume(NSAMP - 1);

    *(float4*)&out[(b * NBOX + nb + g) * SS + c] = acc;
}

extern "C" void kernel_launch(void* const* d_in, const int* in_sizes, int n_in,
                              void* d_out, int out_size, void* d_ws, size_t ws_size,
                              hipStream_t stream) {
    const float* inputs = (const float*)d_in[0];
    const float* boxes  = (const float*)d_in[1];
    float* out = (float*)d_out;
    (void)in_sizes; (void)n_in; (void)out_size; (void)d_ws; (void)ws_size;

    dim3 grid((BATCH * NBOX) / GPB);   // 256 blocks, 4 boxes each
    dim3 block(256);                   // 8 wave32; 2 waves per box (64 lanes x 4 ch)
    psroi_kernel<<<grid, block, 0, stream>>>(inputs, boxes, out);
}